// gnn2rnn_58815282151631
// MI455X (gfx1250) — compile-verified
//
#include <hip/hip_runtime.h>

// Problem constants (match reference)
#define T_    8
#define N_    20000
#define D_    128
#define E_    320000
#define HID_  128
#define H_    128      // LSTM hidden
#define G4_   512      // 4*H
#define OUT_  32
#define LW    8        // waves per block in LSTM kernel

typedef __attribute__((ext_vector_type(2))) float v2f;
typedef __attribute__((ext_vector_type(8))) float v8f;

__device__ __forceinline__ v8f wmma_f32(v2f a, v2f b, v8f c) {
  // D = A(16x4,f32) * B(4x16,f32) + C(16x16,f32)
  return __builtin_amdgcn_wmma_f32_16x16x4_f32(false, a, false, b, (short)0, c,
                                               false, false);
}

__device__ __forceinline__ float sigmoidf_(float x) {
  return 1.0f / (1.0f + __expf(-x));
}

// ---------------- degree ----------------
__global__ void k_degcnt(const int* __restrict__ dst, float* __restrict__ deg) {
  int e = blockIdx.x * blockDim.x + threadIdx.x;
  if (e < E_) atomicAdd(&deg[dst[e]], 1.0f);
}

__global__ void k_deginv(float* __restrict__ deg) {
  int n = blockIdx.x * blockDim.x + threadIdx.x;
  if (n < N_) deg[n] = 1.0f / fmaxf(deg[n], 1.0f);
}

// ---------------- scatter-add aggregation ----------------
// wave per (t,e); lane handles 4 contiguous floats of the 128-wide feature row
__global__ void k_scatter(const float* __restrict__ x, const int* __restrict__ src,
                          const int* __restrict__ dst, float* __restrict__ agg) {
  long long gid = (long long)blockIdx.x * blockDim.x + threadIdx.x;
  int c = (int)(gid & 31);
  long long we = gid >> 5;          // t*E + e
  int t = (int)(we / E_);
  if (t >= T_) return;
  int e = (int)(we - (long long)t * E_);
  int s = src[e], d = dst[e];
  const float4 xv = *(const float4*)(x + ((long long)t * N_ + s) * D_ + c * 4);
  float* ap = agg + ((long long)t * N_ + d) * D_ + c * 4;
  atomicAdd(ap + 0, xv.x);
  atomicAdd(ap + 1, xv.y);
  atomicAdd(ap + 2, xv.z);
  atomicAdd(ap + 3, xv.w);
}

// ---------------- SAGE linear: h = mean@Wl^T + bl + x@Wr^T ----------------
// block = 8 waves; blockIdx.x = 16-row tile of flattened [T*N]; wave = 16-col tile
__global__ __launch_bounds__(256)
void k_sage(const float* __restrict__ agg, const float* __restrict__ x,
            const float* __restrict__ wl, const float* __restrict__ bl,
            const float* __restrict__ wr, const float* __restrict__ dinv,
            float* __restrict__ hfeat) {
  const int lane  = threadIdx.x & 31;
  const int wave  = threadIdx.x >> 5;
  const long row0 = (long)blockIdx.x * 16;
  const int l15   = lane & 15;
  const int khalf = (lane >> 4) << 1;   // 0 or 2 (A/B VGPR layout for f32 K=4)
  const int rbase = (lane >> 4) << 3;   // D layout: lanes>=16 hold M=8..15
  const int j0    = wave * 16;

  const long arow = row0 + l15;
  const float di  = dinv[(int)(arow % N_)];
  const float* aP  = agg + arow * D_;
  const float* xP  = x   + arow * D_;
  const float* wlP = wl + (long)(j0 + l15) * D_;
  const float* wrP = wr + (long)(j0 + l15) * D_;

  v8f acc = (v8f){0.f,0.f,0.f,0.f,0.f,0.f,0.f,0.f};
  for (int k0 = 0; k0 < D_; k0 += 4) {
    int k = k0 + khalf;
    v2f a1; a1.x = aP[k] * di; a1.y = aP[k + 1] * di;
    v2f b1; b1.x = wlP[k];     b1.y = wlP[k + 1];
    acc = wmma_f32(a1, b1, acc);
    v2f a2; a2.x = xP[k];      a2.y = xP[k + 1];
    v2f b2; b2.x = wrP[k];     b2.y = wrP[k + 1];
    acc = wmma_f32(a2, b2, acc);
  }
  const float bias = bl[j0 + l15];
#pragma unroll
  for (int r = 0; r < 8; ++r)
    hfeat[(row0 + rbase + r) * (long)HID_ + j0 + l15] = acc[r] + bias;
}

// ---------------- fused LSTM over all T steps ----------------
// wave owns 16 nodes; c in VGPRs, h in wave-private LDS (ds-ordered recurrence)
__global__ __launch_bounds__(32 * LW, 1)
void k_lstm(const float* __restrict__ hfeat, const float* __restrict__ wih,
            const float* __restrict__ whh, const float* __restrict__ bih,
            const float* __restrict__ bhh, float* __restrict__ out) {
  __shared__ float hsh[LW * 16 * H_];   // 8 waves * 8KB = 64 KB
  const int lane = threadIdx.x & 31;
  const int wave = threadIdx.x >> 5;
  const int tile = blockIdx.x * LW + wave;
  if (tile >= N_ / 16) return;          // wave-uniform: EXEC stays all-ones for WMMA
  const int node0 = tile * 16;
  const int l15   = lane & 15;
  const int khalf = (lane >> 4) << 1;
  const int rbase = (lane >> 4) << 3;
  float* hp = &hsh[wave * 16 * H_];     // h_prev[16][128], wave-private

  for (int i = lane; i < 16 * H_; i += 32) hp[i] = 0.0f;   // h0 = 0

  v8f c[8];
#pragma unroll
  for (int jb = 0; jb < 8; ++jb) c[jb] = (v8f){0.f,0.f,0.f,0.f,0.f,0.f,0.f,0.f};

  // bias (b_ih+b_hh) depends only on the column = lane&15 within a tile,
  // so it is constant across the 8 accumulator VGPRs -> preload into C.
  float bias[4][8];
#pragma unroll
  for (int g = 0; g < 4; ++g)
#pragma unroll
    for (int jb = 0; jb < 8; ++jb) {
      int col = g * H_ + jb * 16 + l15;
      bias[g][jb] = bih[col] + bhh[col];
    }

  for (int t = 0; t < T_; ++t) {
    const float* xrow = hfeat + ((long)t * N_ + node0 + l15) * (long)HID_;
    if (t + 1 < T_)  // pull next step's activations toward the WGP
      __builtin_prefetch(xrow + (long)N_ * HID_, 0, 0);
    const float* hrow = hp + l15 * H_;
    v8f hnew[8];
#pragma unroll
    for (int jb = 0; jb < 8; ++jb) {
      v8f acc[4];
#pragma unroll
      for (int g = 0; g < 4; ++g) {
        float b = bias[g][jb];
        acc[g] = (v8f){b,b,b,b,b,b,b,b};
      }
      for (int k0 = 0; k0 < HID_; k0 += 4) {
        int k = k0 + khalf;
        v2f a;  a.x  = xrow[k]; a.y  = xrow[k + 1];   // input-GEMM A (global)
        v2f ah; ah.x = hrow[k]; ah.y = hrow[k + 1];   // recurrent A (LDS)
#pragma unroll
        for (int g = 0; g < 4; ++g) {
          const float* wi = wih + (long)(g * H_ + jb * 16 + l15) * HID_ + k;
          v2f b1; b1.x = wi[0]; b1.y = wi[1];
          acc[g] = wmma_f32(a, b1, acc[g]);
          const float* wh = whh + (long)(g * H_ + jb * 16 + l15) * H_ + k;
          v2f b2; b2.x = wh[0]; b2.y = wh[1];
          acc[g] = wmma_f32(ah, b2, acc[g]);
        }
      }
#pragma unroll
      for (int r = 0; r < 8; ++r) {        // torch gate order i,f,g,o
        float iv = sigmoidf_(acc[0][r]);
        float fv = sigmoidf_(acc[1][r]);
        float gv = tanhf(acc[2][r]);
        float ov = sigmoidf_(acc[3][r]);
        float cn = fv * c[jb][r] + iv * gv;
        c[jb][r] = cn;
        hnew[jb][r] = ov * tanhf(cn);
      }
    }
    // commit h_t to LDS (all reads of h_{t-1} are done; same-wave DS ops in-order)
#pragma unroll
    for (int jb = 0; jb < 8; ++jb)
#pragma unroll
      for (int r = 0; r < 8; ++r)
        hp[(rbase + r) * H_ + jb * 16 + l15] = hnew[jb][r];
  }

  // out[hor, n, o] = hT[n, hor*32 + o]
  for (int idx = lane; idx < 16 * H_; idx += 32) {
    int m = idx >> 7, col = idx & 127;
    out[(long)(col >> 5) * ((long)N_ * OUT_) + (long)(node0 + m) * OUT_ + (col & 31)]
        = hp[idx];
  }
}

extern "C" void kernel_launch(void* const* d_in, const int* in_sizes, int n_in,
                              void* d_out, int out_size, void* d_ws, size_t ws_size,
                              hipStream_t stream) {
  const float* x    = (const float*)d_in[0];
  const int*   ei   = (const int*)d_in[1];     // [2,E]: row 0 = src, row 1 = dst
  const float* wl   = (const float*)d_in[2];
  const float* blb  = (const float*)d_in[3];
  const float* wr   = (const float*)d_in[4];
  const float* wih  = (const float*)d_in[5];
  const float* whh  = (const float*)d_in[6];
  const float* bih  = (const float*)d_in[7];
  const float* bhh  = (const float*)d_in[8];
  float* out = (float*)d_out;
  (void)in_sizes; (void)n_in; (void)out_size; (void)ws_size;

  const int* srcI = ei;
  const int* dstI = ei + E_;

  // workspace: deg[N] (padded) | agg[T*N*D] | hfeat[T*N*HID]   (~164 MB)
  float* deg   = (float*)d_ws;
  float* agg   = deg + 20480;
  float* hfeat = agg + (size_t)T_ * N_ * D_;

  hipMemsetAsync(deg, 0, (size_t)N_ * sizeof(float), stream);
  hipMemsetAsync(agg, 0, (size_t)T_ * N_ * D_ * sizeof(float), stream);

  k_degcnt<<<(E_ + 255) / 256, 256, 0, stream>>>(dstI, deg);
  k_deginv<<<(N_ + 255) / 256, 256, 0, stream>>>(deg);

  long long sthreads = (long long)T_ * E_ * 32;          // wave per (t,e)
  k_scatter<<<(int)(sthreads / 256), 256, 0, stream>>>(x, srcI, dstI, agg);

  k_sage<<<(T_ * N_) / 16, 256, 0, stream>>>(agg, x, wl, blb, wr, deg, hfeat);

  int ntiles = N_ / 16;                                   // 1250
  k_lstm<<<(ntiles + LW - 1) / LW, 32 * LW, 0, stream>>>(hfeat, wih, whh,
                                                         bih, bhh, out);
}